// CrossEntropyLoss_72851235275272
// MI455X (gfx1250) — compile-verified
//
#include <hip/hip_runtime.h>

typedef float v2f __attribute__((ext_vector_type(2)));
typedef float v8f __attribute__((ext_vector_type(8)));

constexpr int C       = 21;
constexpr int HW      = 512 * 512;          // per-image spatial size
constexpr int NPIX    = 8 * HW;             // 2,097,152 pixels
constexpr int THREADS = 256;                // 8 waves per block (wave32)
constexpr int PPT     = 4;                  // pixels per thread (float4 loads)
constexpr int GROUPS  = NPIX / PPT;         // 524,288 thread work-items
constexpr int BLOCKS  = GROUPS / THREADS;   // 2048 blocks

__global__ __launch_bounds__(THREADS)
void ce_main(const float* __restrict__ pred,
             const long long* __restrict__ tgt,
             float* __restrict__ partials)
{
    const int g  = blockIdx.x * THREADS + threadIdx.x; // group of 4 pixels
    const int p0 = g * PPT;
    const int b  = p0 / HW;
    const int hw = p0 - b * HW;                         // multiple of 4

    // pred[b, c, h, w] ; base points at (b, 0, hw), float4 granularity
    const float4* bp = reinterpret_cast<const float4*>(pred)
                     + ((long)b * C * HW + hw) / 4;

    // Stream all 21 channels for 4 pixels into VGPRs: 21x global_load_b128
    float4 x[C];
#pragma unroll
    for (int c = 0; c < C; ++c) x[c] = bp[c * (HW / 4)];

    // 4 target class ids (int64 in memory, values 0..20)
    const longlong2* t2p = reinterpret_cast<const longlong2*>(tgt);
    longlong2 ta = t2p[2 * g];
    longlong2 tb = t2p[2 * g + 1];
    const int t0 = (int)ta.x, t1 = (int)ta.y, t2 = (int)tb.x, t3 = (int)tb.y;

    // channel max
    float4 mx = x[0];
#pragma unroll
    for (int c = 1; c < C; ++c) {
        mx.x = fmaxf(mx.x, x[c].x);
        mx.y = fmaxf(mx.y, x[c].y);
        mx.z = fmaxf(mx.z, x[c].z);
        mx.w = fmaxf(mx.w, x[c].w);
    }

    // sum of exp(x - max) and select x[target] via cndmask (no dyn indexing)
    float4 s   = make_float4(0.f, 0.f, 0.f, 0.f);
    float4 sel = make_float4(0.f, 0.f, 0.f, 0.f);
#pragma unroll
    for (int c = 0; c < C; ++c) {
        s.x += __expf(x[c].x - mx.x);
        s.y += __expf(x[c].y - mx.y);
        s.z += __expf(x[c].z - mx.z);
        s.w += __expf(x[c].w - mx.w);
        sel.x = (c == t0) ? x[c].x : sel.x;
        sel.y = (c == t1) ? x[c].y : sel.y;
        sel.z = (c == t2) ? x[c].z : sel.z;
        sel.w = (c == t3) ? x[c].w : sel.w;
    }

    float loss = (mx.x + __logf(s.x) - sel.x)
               + (mx.y + __logf(s.y) - sel.y)
               + (mx.z + __logf(s.z) - sel.z)
               + (mx.w + __logf(s.w) - sel.w);

    // ---- Wave32 reduction on the matrix pipe ----
    // A(16x4): lane m (0..15) supplies A[m][0..1] via a.xy, lane 16+m supplies
    // A[m][2..3]. With a=(loss,0) and B=ones(4x16):
    //   D[m][n] = loss_m + loss_{m+16}  (replicated over n)
    v2f a;  a.x  = loss; a.y  = 0.0f;
    v2f bm; bm.x = 1.0f; bm.y = 1.0f;
    v8f acc = {};
    acc = __builtin_amdgcn_wmma_f32_16x16x4_f32(
        /*neg_a=*/false, a, /*neg_b=*/false, bm,
        /*c_mod=*/(short)0, acc, /*reuse_a=*/false, /*reuse_b=*/false);

    // lanes 0-15 hold rows 0-7 in acc[0..7]; lanes 16-31 hold rows 8-15
    float t = acc[0] + acc[1] + acc[2] + acc[3]
            + acc[4] + acc[5] + acc[6] + acc[7];
    float wsum = t + __shfl_xor(t, 16, 32);   // full wave total on every lane

    __shared__ float wpart[THREADS / 32];
    const int lane = threadIdx.x & 31;
    const int wid  = threadIdx.x >> 5;
    if (lane == 0) wpart[wid] = wsum;
    __syncthreads();
    if (threadIdx.x == 0) {
        float bsum = 0.f;
#pragma unroll
        for (int w = 0; w < THREADS / 32; ++w) bsum += wpart[w];
        partials[blockIdx.x] = bsum;
    }
}

__global__ __launch_bounds__(256)
void ce_reduce(const float* __restrict__ partials, float* __restrict__ out)
{
    __shared__ double sd[256];
    double s = 0.0;
    for (int i = threadIdx.x; i < BLOCKS; i += 256) s += (double)partials[i];
    sd[threadIdx.x] = s;
    __syncthreads();
    for (int off = 128; off > 0; off >>= 1) {
        if (threadIdx.x < off) sd[threadIdx.x] += sd[threadIdx.x + off];
        __syncthreads();
    }
    if (threadIdx.x == 0) out[0] = (float)(sd[0] / (double)NPIX);
}

extern "C" void kernel_launch(void* const* d_in, const int* in_sizes, int n_in,
                              void* d_out, int out_size, void* d_ws, size_t ws_size,
                              hipStream_t stream)
{
    const float*     pred = (const float*)d_in[0];
    const long long* tgt  = (const long long*)d_in[1];
    float* partials = (float*)d_ws;           // 2048 floats, fully rewritten each call

    ce_main  <<<BLOCKS, THREADS, 0, stream>>>(pred, tgt, partials);
    ce_reduce<<<1, 256, 0, stream>>>(partials, (float*)d_out);
}